// U3Layer_2379411882434
// MI455X (gfx1250) — compile-verified
//
#include <hip/hip_runtime.h>
#include <hip/hip_bf16.h>

// ---------------------------------------------------------------------------
// 12-qubit U3 layer:  out_b = (A (x) B) @ state_b  with A = kron(g0..g5),
// B = kron(g6..g11)  ==>  reshape state to 64x64:  OUT_b = A * S_b * B^T.
// Two 64x64 complex GEMMs per batch element via V_WMMA_F32_16X16X4_F32,
// using Karatsuba (3 real accumulation chains per complex product) and
// B-operand fragment sharing between each wave's two output tiles.
// ---------------------------------------------------------------------------

typedef __attribute__((ext_vector_type(2))) float v2f;
typedef __attribute__((ext_vector_type(8))) float v8f;

#define WMMA_F32(a, b, c) \
    __builtin_amdgcn_wmma_f32_16x16x4_f32(false, (a), false, (b), (short)0, (c), false, false)

// ---------------- Kernel 1: build A (row-major) and B^T (row-major) --------

__device__ __forceinline__ void gate_elem(const float* __restrict__ t,
                                          int r, int c,
                                          float& re, float& im) {
    const float th = t[0], ph = t[1], la = t[2];
    const float cs = __cosf(th * 0.5f);
    const float sn = __sinf(th * 0.5f);
    if (r == 0) {
        if (c == 0) { re = cs;                im = 0.0f; }
        else        { re = -__cosf(la) * sn;  im = -__sinf(la) * sn; }
    } else {
        if (c == 0) { re = __cosf(ph) * sn;        im = __sinf(ph) * sn; }
        else        { re = __cosf(ph + la) * cs;   im = __sinf(ph + la) * cs; }
    }
}

__global__ __launch_bounds__(256)
void u3_build_factors(const float* __restrict__ thetas,  // [12,3]
                      float* __restrict__ A_re, float* __restrict__ A_im,
                      float* __restrict__ BT_re, float* __restrict__ BT_im) {
    const int t   = blockIdx.x * blockDim.x + threadIdx.x;   // 0..4095
    const int row = t >> 6;
    const int col = t & 63;

    // A element: gates 0..5, gate q uses bit (5-q) of the 6-bit index
    float are = 1.0f, aim = 0.0f;
    #pragma unroll
    for (int q = 0; q < 6; ++q) {
        const int rb = (row >> (5 - q)) & 1;
        const int cb = (col >> (5 - q)) & 1;
        float gre, gim;
        gate_elem(thetas + 3 * q, rb, cb, gre, gim);
        const float nre = are * gre - aim * gim;
        const float nim = are * gim + aim * gre;
        are = nre; aim = nim;
    }
    A_re[t] = are;
    A_im[t] = aim;

    // B[row][col]: gates 6..11; stored transposed: BT[col][row]
    float bre = 1.0f, bim = 0.0f;
    #pragma unroll
    for (int q = 0; q < 6; ++q) {
        const int rb = (row >> (5 - q)) & 1;
        const int cb = (col >> (5 - q)) & 1;
        float gre, gim;
        gate_elem(thetas + 3 * (q + 6), rb, cb, gre, gim);
        const float nre = bre * gre - bim * gim;
        const float nim = bre * gim + bim * gre;
        bre = nre; bim = nim;
    }
    BT_re[col * 64 + row] = bre;
    BT_im[col * 64 + row] = bim;
}

// ---------------- Kernel 2: OUT_b = A * S_b * B^T via f32 WMMA -------------

__global__ __launch_bounds__(256)
void u3_apply(const float* __restrict__ state_re,   // [512,4096]
              const float* __restrict__ state_im,   // [512,4096]
              const float* __restrict__ A_re, const float* __restrict__ A_im,
              const float* __restrict__ BT_re, const float* __restrict__ BT_im,
              float* __restrict__ out)              // [512,4096,2]
{
    __shared__ float Tre[64 * 64];   // T = A * S  (real plane)
    __shared__ float Tim[64 * 64];   // T = A * S  (imag plane)

    const int b    = blockIdx.x;           // batch element
    const int wave = threadIdx.x >> 5;     // 0..7
    const int lane = threadIdx.x & 31;
    const int ll   = lane & 15;            // lane index within half-wave
    const int lh   = lane >> 4;            // 0 = K pair {0,1}, 1 = K pair {2,3}

    // Wave w owns tiles (ti0, tj) and (ti0+2, tj): same column tile, so the
    // B-operand fragments are loaded once and feed both tiles' WMMAs.
    const int tj  = wave & 3;
    const int ti0 = wave >> 2;             // 0 or 1 (second tile is ti0+2)

    const float* __restrict__ Sre = state_re + (size_t)b * 4096;
    const float* __restrict__ Sim = state_im + (size_t)b * 4096;

    // ---------------- Stage 1: T = A * S ----------------
    {
        const int n  = tj * 16 + ll;             // S column for this lane
        const int m0 = (ti0    ) * 16 + ll;      // A row, tile 0
        const int m1 = (ti0 + 2) * 16 + ll;      // A row, tile 1

        // Karatsuba chains: P1 = Ar*Sr, P2 = Ai*Si, P3 = (Ar+Ai)*(Sr+Si)
        v8f p1a = {}, p2a = {}, p3a = {};
        v8f p1b = {}, p2b = {}, p3b = {};

        #pragma unroll 4
        for (int kk = 0; kk < 64; kk += 4) {
            const int k0 = kk + lh * 2;
            // Shared B-operand (S fragment): lane holds S[k0][n], S[k0+1][n]
            v2f sr, si;
            sr.x = Sre[(k0    ) * 64 + n];
            sr.y = Sre[(k0 + 1) * 64 + n];
            si.x = Sim[(k0    ) * 64 + n];
            si.y = Sim[(k0 + 1) * 64 + n];
            const v2f ss = sr + si;

            // Tile 0 A-operand: contiguous K pair -> single b64 load
            v2f ar0 = *(const v2f*)(A_re + m0 * 64 + k0);
            v2f ai0 = *(const v2f*)(A_im + m0 * 64 + k0);
            v2f as0 = ar0 + ai0;
            p1a = WMMA_F32(ar0, sr, p1a);
            p2a = WMMA_F32(ai0, si, p2a);
            p3a = WMMA_F32(as0, ss, p3a);

            // Tile 1 A-operand
            v2f ar1 = *(const v2f*)(A_re + m1 * 64 + k0);
            v2f ai1 = *(const v2f*)(A_im + m1 * 64 + k0);
            v2f as1 = ar1 + ai1;
            p1b = WMMA_F32(ar1, sr, p1b);
            p2b = WMMA_F32(ai1, si, p2b);
            p3b = WMMA_F32(as1, ss, p3b);
        }

        // Combine and stage T in LDS. C/D layout: lanes 0-15 -> M=v,
        // lanes 16-31 -> M=8+v.
        #pragma unroll
        for (int v = 0; v < 8; ++v) {
            const int M = lh * 8 + v;
            const int c = tj * 16 + ll;
            {
                const int idx = ((ti0    ) * 16 + M) * 64 + c;
                const float r = p1a[v] - p2a[v];
                Tre[idx] = r;
                Tim[idx] = p3a[v] - p1a[v] - p2a[v];
            }
            {
                const int idx = ((ti0 + 2) * 16 + M) * 64 + c;
                const float r = p1b[v] - p2b[v];
                Tre[idx] = r;
                Tim[idx] = p3b[v] - p1b[v] - p2b[v];
            }
        }
    }

    __syncthreads();

    // ---------------- Stage 2: OUT = T * B^T ----------------
    {
        const int n  = tj * 16 + ll;             // BT column / OUT column
        const int m0 = (ti0    ) * 16 + ll;      // T row, tile 0
        const int m1 = (ti0 + 2) * 16 + ll;      // T row, tile 1

        v8f p1a = {}, p2a = {}, p3a = {};
        v8f p1b = {}, p2b = {}, p3b = {};

        #pragma unroll 4
        for (int jj = 0; jj < 64; jj += 4) {
            const int k0 = jj + lh * 2;
            // Shared B-operand (BT fragment): coalesced across lanes
            v2f br, bi;
            br.x = BT_re[(k0    ) * 64 + n];
            br.y = BT_re[(k0 + 1) * 64 + n];
            bi.x = BT_im[(k0    ) * 64 + n];
            bi.y = BT_im[(k0 + 1) * 64 + n];
            const v2f bs = br + bi;

            // Tile 0 A-operand: T fragment from LDS (contiguous K pair)
            v2f tr0 = *(const v2f*)(&Tre[m0 * 64 + k0]);
            v2f tf0 = *(const v2f*)(&Tim[m0 * 64 + k0]);
            v2f ts0 = tr0 + tf0;
            p1a = WMMA_F32(tr0, br, p1a);
            p2a = WMMA_F32(tf0, bi, p2a);
            p3a = WMMA_F32(ts0, bs, p3a);

            // Tile 1 A-operand
            v2f tr1 = *(const v2f*)(&Tre[m1 * 64 + k0]);
            v2f tf1 = *(const v2f*)(&Tim[m1 * 64 + k0]);
            v2f ts1 = tr1 + tf1;
            p1b = WMMA_F32(tr1, br, p1b);
            p2b = WMMA_F32(tf1, bi, p2b);
            p3b = WMMA_F32(ts1, bs, p3b);
        }

        float2* __restrict__ outv = (float2*)out;  // (re, im) per element
        #pragma unroll
        for (int v = 0; v < 8; ++v) {
            const int M = lh * 8 + v;
            {
                const size_t idx = (size_t)b * 4096 + ((ti0    ) * 16 + M) * 64 + n;
                outv[idx] = make_float2(p1a[v] - p2a[v],
                                        p3a[v] - p1a[v] - p2a[v]);
            }
            {
                const size_t idx = (size_t)b * 4096 + ((ti0 + 2) * 16 + M) * 64 + n;
                outv[idx] = make_float2(p1b[v] - p2b[v],
                                        p3b[v] - p1b[v] - p2b[v]);
            }
        }
    }
}

// ---------------------------------------------------------------------------

extern "C" void kernel_launch(void* const* d_in, const int* in_sizes, int n_in,
                              void* d_out, int out_size, void* d_ws, size_t ws_size,
                              hipStream_t stream) {
    const float* thetas   = (const float*)d_in[0];   // [12,3]
    const float* state_re = (const float*)d_in[1];   // [512,4096,1]
    const float* state_im = (const float*)d_in[2];   // [512,4096,1]
    float* out = (float*)d_out;                      // [512,4096,2]

    float* A_re  = (float*)d_ws;          // 4096 floats each
    float* A_im  = A_re  + 4096;
    float* BT_re = A_im  + 4096;
    float* BT_im = BT_re + 4096;          // total 64 KB of workspace

    u3_build_factors<<<16, 256, 0, stream>>>(thetas, A_re, A_im, BT_re, BT_im);
    u3_apply<<<512, 256, 0, stream>>>(state_re, state_im,
                                      A_re, A_im, BT_re, BT_im, out);
}